// CRF_7765300871379
// MI455X (gfx1250) — compile-verified
//
#include <hip/hip_runtime.h>
#include <hip/hip_bf16.h>

// Viterbi decode (max-plus semiring + argmax backtrack) for B=1024,T=1024,N=32.
// One wave32 per batch item; lane j == tag j. Backpointers live in LDS.
// CDNA5-specific paths used: global_load_async_to_lds_b128 (ASYNCcnt) +
// s_wait_asynccnt for double-buffered logits staging. WMMA is not applicable:
// (max,+) with argmax is not expressible as D = A*B + C.

#define NSTATE 32
#define WAVES  4      // batches (waves) per workgroup
#define TILE   16     // timesteps per async staging tile

typedef int v4i __attribute__((vector_size(16)));

// ---- CDNA5 async global->LDS copy (16B per lane) ------------------------
// Builtin signature (from hipcc diagnostic): param0 = AS(1) v4i* (global,
// printed as "__device__ *"), param1 = AS(3) v4i* (LDS), then imm offset, cpol.
#if __has_builtin(__builtin_amdgcn_global_load_async_to_lds_b128)
#define ASYNC_CP16(gp, lp)                                                    \
  __builtin_amdgcn_global_load_async_to_lds_b128(                             \
      (__attribute__((address_space(1))) v4i*)(gp),                           \
      (__attribute__((address_space(3))) v4i*)(lp), 0, 0)
#else
#define ASYNC_CP16(gp, lp)                                                    \
  do {                                                                        \
    unsigned _loff = (unsigned)(unsigned long long)                           \
        (__attribute__((address_space(3))) char*)(lp);                        \
    unsigned long long _ga = (unsigned long long)(gp);                        \
    asm volatile("global_load_async_to_lds_b128 %0, %1, off"                  \
                 :: "v"(_loff), "v"(_ga) : "memory");                         \
  } while (0)
#endif

#if __has_builtin(__builtin_amdgcn_s_wait_asynccnt)
#define WAIT_ASYNC(n)                                                         \
  do { __builtin_amdgcn_s_wait_asynccnt(n); asm volatile("" ::: "memory"); } while (0)
#else
#define WAIT_ASYNC(n) asm volatile("s_wait_asynccnt %0" :: "n"(n) : "memory")
#endif

__device__ __forceinline__ float rdlanef(float v, int l) {
  return __int_as_float(__builtin_amdgcn_readlane(__float_as_int(v), l));
}

__launch_bounds__(WAVES * 32, 1)
__global__ void viterbi_crf_kernel(const float* __restrict__ logits,
                                   const float* __restrict__ trans,
                                   const int*   __restrict__ seqlen,
                                   int*         __restrict__ pred,
                                   int T) {
  extern __shared__ char smem[];
  const int lane = threadIdx.x & 31;
  const int wv   = threadIdx.x >> 5;
  const int b    = blockIdx.x * WAVES + wv;

  // LDS layout: [WAVES backpointer slots][WAVES double-buffered logit tiles]
  // bp row stride padded to 257 dwords -> conflict-free banks for 32 lanes.
  const int row_dw  = (T >> 2) + 1;
  const int bp_slot = NSTATE * row_dw * 4;
  char*  bp    = smem + wv * bp_slot;
  float* stage = (float*)(smem + WAVES * bp_slot + wv * (2 * TILE * NSTATE * 4));

  const float* mylog  = logits + (size_t)b * T * NSTATE;
  int*         mypred = pred   + (size_t)b * T;

  int L = seqlen[b];
  if (L > T) L = T;

  // Zero the masked tail t >= L (disjoint from backtrack writes t < L).
  for (int t = L + lane; t < T; t += 32) mypred[t] = 0;

  // Transition column for my tag: tc[i] = trans[i][lane].
  float tc[NSTATE];
#pragma unroll
  for (int i = 0; i < NSTATE; ++i) tc[i] = trans[i * NSTATE + lane];

  float state = mylog[lane];  // t = 0

  if (L > 1) {
    const char* gbase = (const char*)mylog;
    // Prefetch tile 0 (t in [0,16)): 4 x b128 per wave = 16 steps * 128B.
#pragma unroll
    for (int q = 0; q < 4; ++q)
      ASYNC_CP16(gbase + q * 512 + lane * 16,
                 (char*)stage + q * 512 + lane * 16);

    const int ntile = (L + TILE - 1) / TILE;
    unsigned acc = 0;
    int t = 1;
    for (int k = 0; k < ntile; ++k) {
      if (k + 1 < ntile) {
        const char* gs = gbase + (size_t)(k + 1) * (TILE * NSTATE * 4);
        char*       ls = (char*)stage + ((k + 1) & 1) * (TILE * NSTATE * 4);
#pragma unroll
        for (int q = 0; q < 4; ++q)
          ASYNC_CP16(gs + q * 512 + lane * 16, ls + q * 512 + lane * 16);
        WAIT_ASYNC(4);  // in-order completion: tile k is now resident
      } else {
        WAIT_ASYNC(0);
      }

      const float* buf = stage + (k & 1) * (TILE * NSTATE);
      int tend = (k + 1) * TILE;
      if (tend > L) tend = L;
      for (; t < tend; ++t) {
        float lg = buf[(t & (TILE - 1)) * NSTATE + lane];  // conflict-free
        // max-plus column reduce with first-max argmax (matches jnp.argmax)
        float best = tc[0] + rdlanef(state, 0);
        int   bpi  = 0;
#pragma unroll
        for (int i = 1; i < NSTATE; ++i) {
          float c = tc[i] + rdlanef(state, i);
          bool  g = c > best;
          bpi  = g ? i : bpi;
          best = g ? c : best;
        }
        state = best + lg;
        // pack 4 backpointers per dword; rewrite-in-place keeps it branchless
        unsigned sh = (unsigned)(t & 3) * 8u;
        acc = (((t & 3) == 0) ? 0u : acc) | ((unsigned)bpi << sh);
        *(unsigned*)(bp + (size_t)(lane * row_dw + (t >> 2)) * 4) = acc;
      }
    }
  }

  // Wave-wide argmax of final state (first-max tie-break on smaller index).
  float val = state;
  int   idx = lane;
#pragma unroll
  for (int m = 16; m >= 1; m >>= 1) {
    float ov = __shfl_xor(val, m, 32);
    int   oi = __shfl_xor(idx, m, 32);
    bool take = (ov > val) || (ov == val && oi < idx);
    val = take ? ov : val;
    idx = take ? oi : idx;
  }

  // Sequential backtrack through LDS backpointers (lane 0; ~30cy/hop in LDS).
  if (lane == 0 && L > 0) {
    int tag = idx;
    for (int t = L - 1; t >= 1; --t) {
      mypred[t] = tag;
      tag = *(const unsigned char*)(bp + (size_t)tag * row_dw * 4 + t);
    }
    mypred[0] = tag;
  }
}

extern "C" void kernel_launch(void* const* d_in, const int* in_sizes, int n_in,
                              void* d_out, int out_size, void* d_ws, size_t ws_size,
                              hipStream_t stream) {
  (void)n_in; (void)out_size; (void)d_ws; (void)ws_size;
  const float* logits = (const float*)d_in[0];
  const float* trans  = (const float*)d_in[1];
  const int*   seqlen = (const int*)d_in[2];
  int*         pred   = (int*)d_out;  // reference output dtype is int32

  const int B = in_sizes[2];
  const int T = in_sizes[0] / (B * NSTATE);

  const int blocks = (B + WAVES - 1) / WAVES;
  const size_t row_dw  = (size_t)(T >> 2) + 1;
  const size_t shmem   = (size_t)WAVES * (NSTATE * row_dw * 4)   // backpointers
                       + (size_t)WAVES * (2 * TILE * NSTATE * 4); // logit stage

  viterbi_crf_kernel<<<blocks, WAVES * 32, shmem, stream>>>(
      logits, trans, seqlen, pred, T);
}